// LTIStagedRouter_48352741818705
// MI455X (gfx1250) — compile-verified
//
#include <hip/hip_runtime.h>

#define N_CLUSTERS 8
#define N_C        6144
#define T_LEN      2048
#define D_TAPS     32
#define K_TR       2048
#define PAD_F      32                      // leading zero floats per padded row
#define STRIDE_P   (T_LEN + PAD_F)         // 2080 floats; 8320B = 128B aligned

typedef __attribute__((ext_vector_type(2)))  float v2f;
typedef __attribute__((ext_vector_type(4)))  float v4f;
typedef __attribute__((ext_vector_type(8)))  float v8f;

// ---------------------------------------------------------------------------
// Kernel 1: runoff row = [ 32 zeros | x_c row ].  x is streamed once ->
// non-temporal load so the 50MB slab does not evict L2-resident runoff / y.
// STRIDE4/PAD4 in float4 units.  Grid covers N_C * STRIDE4 threads exactly.
// ---------------------------------------------------------------------------
template <int STRIDE4, int PAD4>
__global__ void lti_copy_slab(const float* __restrict__ x_c,
                              float* __restrict__ runoff) {
    size_t i    = (size_t)blockIdx.x * blockDim.x + threadIdx.x;
    int    row  = (int)(i / STRIDE4);
    int    col4 = (int)(i % STRIDE4);
    v4f v = (v4f)0.0f;
    if (col4 >= PAD4)
        v = __builtin_nontemporal_load(
                (const v4f*)(x_c + (size_t)row * T_LEN) + (col4 - PAD4));
    ((v4f*)(runoff + (size_t)row * (STRIDE4 * 4)))[col4] = v;
}

// ---------------------------------------------------------------------------
// Kernel 2: runoff[dst_c[j], :] += y_prev[src_prev[j], :]
// dst indices may collide -> f32 atomic add (native global_atomic_add_f32).
// ---------------------------------------------------------------------------
__global__ void lti_scatter_add(const float* __restrict__ y_prev,
                                const int* __restrict__ dst_c,
                                const int* __restrict__ src_prev,
                                float* __restrict__ runoff,
                                int stride, int pad) {
    size_t i  = (size_t)blockIdx.x * blockDim.x + threadIdx.x;  // over K*T/4
    int    j  = (int)(i >> 9);          // T/4 = 512
    int    t4 = (int)(i & 511);
    int    s  = src_prev[j];
    int    d  = dst_c[j];
    v4f    v  = ((const v4f*)(y_prev + (size_t)s * T_LEN))[t4];
    float* o  = runoff + (size_t)d * stride + pad + 4 * (size_t)t4;
    atomicAdd(o + 0, v[0]);
    atomicAdd(o + 1, v[1]);
    atomicAdd(o + 2, v[2]);
    atomicAdd(o + 3, v[3]);
}

// ---------------------------------------------------------------------------
// Kernel 3: per-node causal FIR on the matrix units, full fp32.
// One wave32 per node; per 256-output tile:
//   Y(16x16) = A2*R'' + A1*R' + A0*R   as 12 chained V_WMMA_F32_16X16X4_F32.
// GUARD=false (padded rows): every B load is unconditional with an immediate
// offset from one per-lane base -> clause-batched, pipelined across tiles.
// ---------------------------------------------------------------------------
template <int STRIDE, int PAD, bool GUARD>
__global__ void __launch_bounds__(256)
lti_fir_wmma(const float* __restrict__ runoff,
             const float* __restrict__ taps_c,   // kernels + c*N_C*D
             float* __restrict__ y_out) {        // d_out  + c*N_C*T
    const int lane = threadIdx.x & 31;
    const int wave = threadIdx.x >> 5;
    const int node = blockIdx.x * 8 + wave;
    const int l16  = lane & 15;     // M for A, N for B/D
    const int hi   = lane >> 4;

    const float tap = taps_c[(size_t)node * D_TAPS + lane];  // tap[lane]

    // A slices (32-bit 16x4 A layout): lane -> M=l16, VGPR v -> u=4q+2hi+v.
    // matrix m: shift = -32+16m, A_m[j,u] = tap[j-u+32-16m] if 0<=idx<=31.
    v2f a[3][4];
#pragma unroll
    for (int m = 0; m < 3; ++m)
#pragma unroll
        for (int q = 0; q < 4; ++q)
#pragma unroll
            for (int v = 0; v < 2; ++v) {
                int u      = 4 * q + 2 * hi + v;
                int tapidx = l16 - u + 32 - 16 * m;
                float tv   = __shfl(tap, tapidx & 31, 32);
                a[m][q][v] = (tapidx >= 0 && tapidx <= 31) ? tv : 0.0f;
            }

    // B (4x16, 32-bit): lane -> N=l16, VGPR v -> row u=4q+2hi+v of R_m:
    // two consecutive floats at r[t0 + 16*l16 + (-32+16m) + 4q + 2hi].
    const int    bcol  = 16 * l16 + 2 * hi;
    const float* lbase = runoff + (size_t)node * STRIDE + (PAD - 32) + bcol;
    float*       yrow  = y_out  + (size_t)node * T_LEN;

    for (int t0 = 0; t0 < T_LEN; t0 += 256) {
        // gather all 12 B operands first (independent -> one load clause)
        v2f b[3][4];
#pragma unroll
        for (int m = 0; m < 3; ++m)
#pragma unroll
            for (int q = 0; q < 4; ++q) {
                const int off = t0 + 16 * m + 4 * q;
                if (GUARD) {
                    int idx = bcol - 32 + t0 + 16 * m + 4 * q;
                    b[m][q] = (idx >= 0) ? *(const v2f*)(lbase + off)
                                         : (v2f)0.0f;
                } else {
                    b[m][q] = *(const v2f*)(lbase + off);
                }
            }
        v8f acc = {};
#pragma unroll
        for (int m = 0; m < 3; ++m)
#pragma unroll
            for (int q = 0; q < 4; ++q)
                acc = __builtin_amdgcn_wmma_f32_16x16x4_f32(
                          false, a[m][q], false, b[m][q], (short)0, acc,
                          false, false);
        // D layout: lane, vgpr v -> y[t0 + 16*l16 + 8*hi + v] (contiguous)
        float* op = yrow + t0 + 16 * l16 + 8 * hi;
        *(v4f*)(op)     = (v4f){acc[0], acc[1], acc[2], acc[3]};
        *(v4f*)(op + 4) = (v4f){acc[4], acc[5], acc[6], acc[7]};
    }
}

// ---------------------------------------------------------------------------
// Host launcher: 8 sequential cluster stages on `stream`.
// d_ws holds the runoff buffer; padded layout (6144 x 2080 f32 = 51.1MB) if
// ws_size allows, else guarded unpadded layout (48MB).
// ---------------------------------------------------------------------------
extern "C" void kernel_launch(void* const* d_in, const int* in_sizes, int n_in,
                              void* d_out, int out_size, void* d_ws, size_t ws_size,
                              hipStream_t stream) {
    const float* x       = (const float*)d_in[0];
    const float* kernels = (const float*)d_in[1];
    const int*   dst_idx = (const int*)d_in[2];
    const int*   src_idx = (const int*)d_in[3];
    // d_in[4] = g_in, d_in[5] = g_out: contiguous slot groups rolled by one
    // cluster -> bucket_in[j] == y_prev[src_prev[j]], never materialized.
    float* out    = (float*)d_out;
    float* runoff = (float*)d_ws;

    const bool padded = ws_size >= (size_t)N_C * STRIDE_P * sizeof(float);

    const int copy_blocks_p  = (N_C * (STRIDE_P / 4)) / 256;  // 12480
    const int copy_blocks_u  = (N_C * (T_LEN   / 4)) / 256;   // 12288
    const int scatter_blocks = (K_TR * T_LEN / 4) / 256;      //  4096
    const int fir_blocks     = N_C / 8;                       //   768

    for (int c = 0; c < N_CLUSTERS; ++c) {
        if (padded)
            lti_copy_slab<STRIDE_P / 4, PAD_F / 4><<<copy_blocks_p, 256, 0, stream>>>(
                x + (size_t)c * N_C * T_LEN, runoff);
        else
            lti_copy_slab<T_LEN / 4, 0><<<copy_blocks_u, 256, 0, stream>>>(
                x + (size_t)c * N_C * T_LEN, runoff);

        if (c > 0) {
            lti_scatter_add<<<scatter_blocks, 256, 0, stream>>>(
                out + (size_t)(c - 1) * N_C * T_LEN,
                dst_idx + (size_t)c * K_TR,
                src_idx + (size_t)(c - 1) * K_TR,
                runoff,
                padded ? STRIDE_P : T_LEN,
                padded ? PAD_F : 0);
        }

        if (padded)
            lti_fir_wmma<STRIDE_P, PAD_F, false><<<fir_blocks, 256, 0, stream>>>(
                runoff, kernels + (size_t)c * N_C * D_TAPS,
                out + (size_t)c * N_C * T_LEN);
        else
            lti_fir_wmma<T_LEN, 0, true><<<fir_blocks, 256, 0, stream>>>(
                runoff, kernels + (size_t)c * N_C * D_TAPS,
                out + (size_t)c * N_C * T_LEN);
    }
}